// ToRGBLayer_66735201845785
// MI455X (gfx1250) — compile-verified
//
#include <hip/hip_runtime.h>

typedef __attribute__((ext_vector_type(2))) float v2f;
typedef __attribute__((ext_vector_type(8))) float v8f;

#define B_     16
#define CIN_   512
#define WDIM_  512
#define HW_    (128 * 128)
#define NT     4                       // 16-pixel N-tiles per wave
#define GAIN_  0.04419417382415922f    // 1/sqrt(512), both FC_GAIN and WEIGHT_GAIN
#define CLAMP_ 256.0f

// ---------------------------------------------------------------------------
// Kernel 1: styles + fused modulated weight table.
// wmod4[b][c][o] = conv_w[o][c] * styles[b][c]  (o=0..2), wmod4[b][c][3] = 0
// One thread per (b, c): three 512-length dot products. ~25 MFLOP total.
// ---------------------------------------------------------------------------
__global__ void tor_style_kernel(const float* __restrict__ w,
                                 const float* __restrict__ aW,
                                 const float* __restrict__ ab,
                                 const float* __restrict__ cw,
                                 float* __restrict__ wmod4) {
    const int idx = blockIdx.x * blockDim.x + threadIdx.x;   // exact grid: B_*CIN_
    const int b = idx >> 9;
    const int c = idx & (CIN_ - 1);

    const float4* wr = (const float4*)(w + (size_t)b * WDIM_);
    const float4* W0 = (const float4*)(aW + (size_t)(c)            * WDIM_);
    const float4* W1 = (const float4*)(aW + (size_t)(c + CIN_)     * WDIM_);
    const float4* W2 = (const float4*)(aW + (size_t)(c + 2 * CIN_) * WDIM_);

    float s0 = 0.f, s1 = 0.f, s2 = 0.f;
    for (int k = 0; k < WDIM_ / 4; ++k) {
        float4 wv = wr[k];
        float4 a0 = W0[k];
        float4 a1 = W1[k];
        float4 a2 = W2[k];
        s0 += wv.x * a0.x + wv.y * a0.y + wv.z * a0.z + wv.w * a0.w;
        s1 += wv.x * a1.x + wv.y * a1.y + wv.z * a1.z + wv.w * a1.w;
        s2 += wv.x * a2.x + wv.y * a2.y + wv.z * a2.z + wv.w * a2.w;
    }
    s0 = s0 * GAIN_ + ab[c];
    s1 = s1 * GAIN_ + ab[c + CIN_];
    s2 = s2 * GAIN_ + ab[c + 2 * CIN_];
    const float style = (s0 * s1 + s2) * GAIN_;

    float* o = wmod4 + ((size_t)b * CIN_ + c) * 4;
    o[0] = cw[0 * CIN_ + c] * style;
    o[1] = cw[1 * CIN_ + c] * style;
    o[2] = cw[2 * CIN_ + c] * style;
    o[3] = 0.f;   // zero pad -> A-matrix rows M>=3 read this, giving zero rows
}

// ---------------------------------------------------------------------------
// Kernel 2: modulated 1x1 conv via V_WMMA_F32_16X16X4_F32.
// Per b: [3x512] @ [512x16384]. HBM-bound: 512 MB of x streamed once (NT hint).
// A tile (16x4): lane L -> M = L&15, K = 2*(L>>4)+v; rows M>=3 are zero.
// B tile (4x16): lane L -> N = L&15, K = 2*(L>>4)+v  (same K expr as A).
// C/D (16x16): VGPR r, lanes 0-15 -> M=r, so channel o lives in acc[o], lanes 0-15.
// ---------------------------------------------------------------------------
__global__ void __launch_bounds__(256, 1)
tor_conv_kernel(const float* __restrict__ x,
                const float* __restrict__ wmod4,
                const float* __restrict__ cb,
                float* __restrict__ out) {
    const int lane = threadIdx.x & 31;
    const int wave = (blockIdx.x * blockDim.x + threadIdx.x) >> 5;

    const int jobsPerB = HW_ / (16 * NT);               // 256
    const int b        = wave / jobsPerB;
    const int pix0     = (wave % jobsPerB) * (16 * NT) + (lane & 15);
    const int krow     = (lane >> 4) << 1;              // 0 or 2
    const int m        = lane & 15;
    const int comp     = (m < 3) ? m : 3;               // pad component -> zero rows

    const float* xb = x + (size_t)b * CIN_ * HW_;
    const float* wb = wmod4 + (size_t)b * CIN_ * 4;

    v8f acc[NT] = {};

#pragma unroll 2
    for (int c = 0; c < CIN_; c += 4) {
        const int k0 = c + krow;
        v2f a;
        a.x = wb[(k0 + 0) * 4 + comp];                  // cached, tiny footprint
        a.y = wb[(k0 + 1) * 4 + comp];
        const float* xr0 = xb + (size_t)(k0 + 0) * HW_ + pix0;
        const float* xr1 = xb + (size_t)(k0 + 1) * HW_ + pix0;
#pragma unroll
        for (int t = 0; t < NT; ++t) {
            v2f bm;
            bm.x = __builtin_nontemporal_load(xr0 + t * 16);   // streamed once
            bm.y = __builtin_nontemporal_load(xr1 + t * 16);
            acc[t] = __builtin_amdgcn_wmma_f32_16x16x4_f32(
                         false, a, false, bm, (short)0, acc[t], false, false);
        }
    }

    if (lane < 16) {                                    // lanes 0-15 hold M=0..7
        const float b0 = cb[0], b1 = cb[1], b2 = cb[2];
#pragma unroll
        for (int t = 0; t < NT; ++t) {
            const int p = pix0 + t * 16;                // pix0 already includes lane
            float v0 = fminf(fmaxf(acc[t][0] + b0, -CLAMP_), CLAMP_);
            float v1 = fminf(fmaxf(acc[t][1] + b1, -CLAMP_), CLAMP_);
            float v2 = fminf(fmaxf(acc[t][2] + b2, -CLAMP_), CLAMP_);
            __builtin_nontemporal_store(v0, out + ((size_t)b * 3 + 0) * HW_ + p);
            __builtin_nontemporal_store(v1, out + ((size_t)b * 3 + 1) * HW_ + p);
            __builtin_nontemporal_store(v2, out + ((size_t)b * 3 + 2) * HW_ + p);
        }
    }
}

extern "C" void kernel_launch(void* const* d_in, const int* in_sizes, int n_in,
                              void* d_out, int out_size, void* d_ws, size_t ws_size,
                              hipStream_t stream) {
    const float* x   = (const float*)d_in[0];   // [16,512,128,128]
    const float* w   = (const float*)d_in[1];   // [16,512]
    const float* aW  = (const float*)d_in[2];   // [1536,512]
    const float* ab  = (const float*)d_in[3];   // [1536]
    const float* cw  = (const float*)d_in[4];   // [3,512]
    const float* cbv = (const float*)d_in[5];   // [3]
    float* out   = (float*)d_out;               // [16,3,128,128]
    float* wmod4 = (float*)d_ws;                // [16,512,4] = 128 KB

    // Kernel 1: exact grid of B_*CIN_ = 8192 threads.
    tor_style_kernel<<<(B_ * CIN_) / 256, 256, 0, stream>>>(w, aW, ab, cw, wmod4);

    // Kernel 2: 16 b * 256 jobs = 4096 waves; 8 waves (256 thr) per block.
    const int jobs = B_ * (HW_ / (16 * NT));    // 4096
    tor_conv_kernel<<<jobs / 8, 256, 0, stream>>>(x, wmod4, cbv, out);
}